// DifferentiableQUBO_57947698757715
// MI455X (gfx1250) — compile-verified
//
#include <hip/hip_runtime.h>

// ---------------------------------------------------------------------------
// DifferentiableQUBO: decisions = sigmoid(scores); loss = sum(decisions^2)
//
// Bandwidth-bound: 128 MB of traffic @ 23.3 TB/s ~ 5.5 us floor. Strategy:
//   * B128 non-temporal loads/stores (no reuse -> don't pollute L2)
//   * v_exp_f32 + v_rcp_f32 fast sigmoid (TRANS pipe, free vs memory)
//   * wave32 shfl reduction -> LDS -> per-block partial in d_ws (deterministic)
//   * final 1-wave kernel: 8 independent B128 loads (one latency, not 32
//     serialized b32 round-trips), then V_WMMA_F32_16X16X4_F32 folds the
//     32 lane accumulators to a scalar.
// ---------------------------------------------------------------------------

typedef float f32x4 __attribute__((ext_vector_type(4)));
typedef float v2f   __attribute__((ext_vector_type(2)));
typedef float v8f   __attribute__((ext_vector_type(8)));

constexpr int TPB  = 256;   // 8 waves per block (wave32)
constexpr int NBLK = 1024;  // partials = 4 KB in d_ws (256 float4s)

__device__ __forceinline__ float fast_sigmoid(float x) {
  // sigmoid(x) = 1 / (1 + 2^(-x * log2(e)))  -> v_exp_f32 + v_rcp_f32
  float e = __builtin_amdgcn_exp2f(-1.44269504088896340736f * x);
  return __builtin_amdgcn_rcpf(1.0f + e);
}

__global__ void qubo_sigmoid_kernel(const float* __restrict__ scores,
                                    float* __restrict__ decisions,
                                    float* __restrict__ partials,
                                    int n) {
  const int stride = gridDim.x * blockDim.x;
  const int tid    = blockIdx.x * blockDim.x + threadIdx.x;
  const int nvec   = n >> 2;

  const f32x4* in4  = reinterpret_cast<const f32x4*>(scores);
  f32x4*       out4 = reinterpret_cast<f32x4*>(decisions);

  float acc = 0.0f;

  // Main vectorized stream: global_load_b128 (NT) -> sigmoid -> b128 NT store
  for (int i = tid; i < nvec; i += stride) {
    f32x4 v = __builtin_nontemporal_load(in4 + i);
    f32x4 d;
    d[0] = fast_sigmoid(v[0]);
    d[1] = fast_sigmoid(v[1]);
    d[2] = fast_sigmoid(v[2]);
    d[3] = fast_sigmoid(v[3]);
    acc = fmaf(d[0], d[0], acc);
    acc = fmaf(d[1], d[1], acc);
    acc = fmaf(d[2], d[2], acc);
    acc = fmaf(d[3], d[3], acc);
    __builtin_nontemporal_store(d, out4 + i);
  }

  // Scalar tail (empty for N = 2^24, kept for safety)
  for (int i = (nvec << 2) + tid; i < n; i += stride) {
    float d = fast_sigmoid(scores[i]);
    decisions[i] = d;
    acc = fmaf(d, d, acc);
  }

  // Wave32 reduction (5 shfl_xor steps)
  for (int off = 16; off > 0; off >>= 1)
    acc += __shfl_xor(acc, off, 32);

  __shared__ float smem[TPB / 32];
  const int lane = threadIdx.x & 31;
  const int wid  = threadIdx.x >> 5;
  if (lane == 0) smem[wid] = acc;
  __syncthreads();

  if (wid == 0) {
    float v = (lane < (TPB / 32)) ? smem[lane] : 0.0f;
    for (int off = (TPB / 64); off > 0; off >>= 1)  // 8 values: 4,2,1
      v += __shfl_xor(v, off, 32);
    if (lane == 0) partials[blockIdx.x] = v;  // overwritten every call
  }
}

// One wave (EXEC all-ones, as WMMA requires). NBLK partials = VEC_PER_LANE
// independent B128 loads per lane (issued back-to-back, single wait), then
// the 32-lane -> scalar step uses V_WMMA_F32_16X16X4_F32:
//   A (16x4 f32): lanes 0-15 put acc at K=0 (VGPR0), lanes 16-31 at K=2;
//   VGPR1 (K=1/K=3) = 0.  B = ones  =>  D[m][n] = acc_m + acc_{m+16}.
//   Lane-local sum of the 8 C/D VGPRs gives half-sums; shfl_xor(16) merges.
__global__ void qubo_final_reduce(const float* __restrict__ partials,
                                  float* __restrict__ loss_out) {
  constexpr int VEC_PER_LANE = NBLK / (32 * 4);  // 8
  static_assert(NBLK % 128 == 0, "NBLK must be a multiple of 128");

  const f32x4* p4 = reinterpret_cast<const f32x4*>(partials);

  // Independent loads: no accumulator in the address/data chain, so the
  // compiler issues all 8 global_load_b128 then waits once.
  f32x4 v[VEC_PER_LANE];
#pragma unroll
  for (int j = 0; j < VEC_PER_LANE; ++j)
    v[j] = p4[threadIdx.x + 32 * j];

  float acc = 0.0f;
#pragma unroll
  for (int j = 0; j < VEC_PER_LANE; ++j)
    acc += (v[j][0] + v[j][1]) + (v[j][2] + v[j][3]);

#if __has_builtin(__builtin_amdgcn_wmma_f32_16x16x4_f32)
  v2f a; a[0] = acc;  a[1] = 0.0f;
  v2f b; b[0] = 1.0f; b[1] = 1.0f;
  v8f c = {};
  c = __builtin_amdgcn_wmma_f32_16x16x4_f32(
      /*neg_a=*/false, a, /*neg_b=*/false, b,
      /*c_mod=*/(short)0, c, /*reuse_a=*/false, /*reuse_b=*/false);
  float half_sum = ((c[0] + c[1]) + (c[2] + c[3])) +
                   ((c[4] + c[5]) + (c[6] + c[7]));
  float total = half_sum + __shfl_xor(half_sum, 16, 32);
#else
  for (int off = 16; off > 0; off >>= 1)
    acc += __shfl_xor(acc, off, 32);
  float total = acc;
#endif

  if (threadIdx.x == 0)
    *loss_out = total;  // direct overwrite: no zero-init, deterministic
}

extern "C" void kernel_launch(void* const* d_in, const int* in_sizes, int n_in,
                              void* d_out, int out_size, void* d_ws, size_t ws_size,
                              hipStream_t stream) {
  const float* scores = (const float*)d_in[0];
  float* out = (float*)d_out;            // [0, n): decisions; [n]: qubo_loss
  const int n = in_sizes[0];
  float* partials = (float*)d_ws;        // NBLK floats = 4 KB scratch

  qubo_sigmoid_kernel<<<NBLK, TPB, 0, stream>>>(scores, out, partials, n);
  qubo_final_reduce<<<1, 32, 0, stream>>>(partials, out + n);
}